// PillarFeatureNet_65128884076643
// MI455X (gfx1250) — compile-verified
//
#include <hip/hip_runtime.h>
#include <hip/hip_bf16.h>
#include <math.h>

typedef __attribute__((ext_vector_type(2))) float v2f;
typedef __attribute__((ext_vector_type(8))) float v8f;

#define PFN_P    32
#define PFN_INC  9
#define PFN_OUTC 64

__device__ __forceinline__ float wave_sum32(float x) {
    #pragma unroll
    for (int off = 16; off > 0; off >>= 1)
        x += __shfl_xor(x, off, 32);
    return x;
}

// ---------------------------------------------------------------------------
// Kernel 0: zero the stats accumulator region of the workspace (64 floats).
// ---------------------------------------------------------------------------
__global__ void pfn_zero(float* __restrict__ S) {
    if (threadIdx.x < 64) S[threadIdx.x] = 0.0f;
}

// ---------------------------------------------------------------------------
// Kernel 1: accumulate Sum(f) (9) and Sum(f f^T) (45 unique) over all M*P
// points. One wave per pillar (grid-stride); per-lane register accumulators,
// wave shuffle-reduce at the end, float atomics into workspace.
// S layout: [0..8] = Sum f ; [16..60] = upper-triangular Sum f_i f_j.
// ---------------------------------------------------------------------------
__global__ __launch_bounds__(256) void pfn_stats(const float* __restrict__ voxels,
                                                 const int* __restrict__ coords,
                                                 const int* __restrict__ nump,
                                                 float* __restrict__ S, int Mn) {
    const int lane = threadIdx.x & 31;
    const int gw   = (int)((blockIdx.x * blockDim.x + threadIdx.x) >> 5);
    const int nw   = (int)((gridDim.x * blockDim.x) >> 5);

    float s1[9];
    float s2[45];
    #pragma unroll
    for (int i = 0; i < 9; ++i) s1[i] = 0.0f;
    #pragma unroll
    for (int i = 0; i < 45; ++i) s2[i] = 0.0f;

    for (int m = gw; m < Mn; m += nw) {
        const float4 v = *(const float4*)(voxels + (size_t)m * (PFN_P * 4) + lane * 4);
        const float sx = wave_sum32(v.x);
        const float sy = wave_sum32(v.y);
        const float sz = wave_sum32(v.z);
        int np = nump[m]; if (np < 1) np = 1;
        const float inv = 1.0f / (float)np;
        const float mx = sx * inv, my = sy * inv, mz = sz * inv;
        const float px = (float)coords[m * 4 + 3] * 0.2f - 51.2f;
        const float py = (float)coords[m * 4 + 2] * 0.2f - 51.2f;

        float f[9];
        f[0] = v.x; f[1] = v.y; f[2] = v.z; f[3] = v.w;
        f[4] = v.x - mx; f[5] = v.y - my; f[6] = v.z - mz;
        f[7] = v.x - px; f[8] = v.y - py;

        #pragma unroll
        for (int i = 0; i < 9; ++i) s1[i] += f[i];
        int idx = 0;
        #pragma unroll
        for (int i = 0; i < 9; ++i)
            #pragma unroll
            for (int j = i; j < 9; ++j) { s2[idx] += f[i] * f[j]; ++idx; }
    }

    #pragma unroll
    for (int i = 0; i < 9; ++i) {
        const float t = wave_sum32(s1[i]);
        if (lane == 0) atomicAdd(&S[i], t);
    }
    #pragma unroll
    for (int i = 0; i < 45; ++i) {
        const float t = wave_sum32(s2[i]);
        if (lane == 0) atomicAdd(&S[16 + i], t);
    }
}

// ---------------------------------------------------------------------------
// Kernel 2: fold BN into the GEMM. For each output channel o:
//   mean_o = (S1/N) . W[:,o]
//   var_o  = w^T (S2/N) w - mean_o^2
//   scale  = gamma * rsqrt(var + eps);  bias_o = beta - mean*scale
// Writes Wscaled (12x64, rows 9..11 zero) and bias (64) into workspace.
// ---------------------------------------------------------------------------
__global__ void pfn_bn(const float* __restrict__ S, const float* __restrict__ W,
                       const float* __restrict__ gamma, const float* __restrict__ beta,
                       float* __restrict__ Ws, float* __restrict__ bias, int Mn) {
    const int o = threadIdx.x;            // 64 threads
    const float invN = 1.0f / ((float)Mn * (float)PFN_P);

    float w[9];
    #pragma unroll
    for (int c = 0; c < 9; ++c) w[c] = W[c * PFN_OUTC + o];

    float mean = 0.0f;
    #pragma unroll
    for (int c = 0; c < 9; ++c) mean += S[c] * w[c];
    mean *= invN;

    float e2 = 0.0f;
    int idx = 0;
    #pragma unroll
    for (int i = 0; i < 9; ++i)
        #pragma unroll
        for (int j = i; j < 9; ++j) {
            const float t = S[16 + idx] * w[i] * w[j];
            e2 += (i == j) ? t : 2.0f * t;
            ++idx;
        }
    e2 *= invN;

    const float var = e2 - mean * mean;
    const float sc  = gamma[o] * rsqrtf(var + 1e-3f);
    bias[o] = beta[o] - mean * sc;

    #pragma unroll
    for (int k = 0; k < 12; ++k)
        Ws[k * PFN_OUTC + o] = (k < 9) ? W[k * PFN_OUTC + o] * sc : 0.0f;
}

// ---------------------------------------------------------------------------
// Kernel 3: main fused pass. One wave per pillar.
//   - lane p loads point p's voxel, wave-reduce xyz mean, build 12 features
//     (3 zero-pad) into LDS
//   - 32x12 @ 12x64 GEMM via v_wmma_f32_16x16x4_f32: 2 M-tiles x 4 N-tiles x
//     3 K-steps (A/B layouts per ISA 7.12.2)
//   - epilogue: max over rows (8 regs, +cross-tile, +shfl_xor 16), add bias,
//     relu, store (relu/max commute; bias is per-column)
// ---------------------------------------------------------------------------
__global__ __launch_bounds__(256) void pfn_main(const float* __restrict__ voxels,
                                                const int* __restrict__ coords,
                                                const int* __restrict__ nump,
                                                const float* __restrict__ Ws,
                                                const float* __restrict__ bias,
                                                float* __restrict__ out, int Mn) {
    __shared__ float fsh[8][32][12];

    const int lane = threadIdx.x & 31;
    const int wv   = threadIdx.x >> 5;
    const int m    = blockIdx.x * 8 + wv;
    const bool active = (m < Mn);
    const int  mm  = active ? m : 0;

    // ---- feature build ----
    const float4 v = *(const float4*)(voxels + (size_t)mm * (PFN_P * 4) + lane * 4);
    const float sx = wave_sum32(v.x);
    const float sy = wave_sum32(v.y);
    const float sz = wave_sum32(v.z);
    int np = nump[mm]; if (np < 1) np = 1;
    const float inv = 1.0f / (float)np;
    const float mx = sx * inv, my = sy * inv, mz = sz * inv;
    const float px = (float)coords[mm * 4 + 3] * 0.2f - 51.2f;
    const float py = (float)coords[mm * 4 + 2] * 0.2f - 51.2f;

    float* fs = &fsh[wv][lane][0];
    fs[0] = v.x; fs[1] = v.y; fs[2] = v.z; fs[3] = v.w;
    fs[4] = v.x - mx; fs[5] = v.y - my; fs[6] = v.z - mz;
    fs[7] = v.x - px; fs[8] = v.y - py;
    fs[9] = 0.0f; fs[10] = 0.0f; fs[11] = 0.0f;
    __syncthreads();

    const int nlo   = lane & 15;
    const int khalf = (lane >> 4) << 1;   // lanes 0-15 -> K+0/K+1, lanes 16-31 -> K+2/K+3

    // ---- B fragments (4x16 per tile), from BN-folded W in workspace ----
    v2f B[3][4];
    #pragma unroll
    for (int kt = 0; kt < 3; ++kt) {
        const int kk = 4 * kt + khalf;
        #pragma unroll
        for (int nt = 0; nt < 4; ++nt) {
            const int n = nlo + 16 * nt;
            B[kt][nt].x = Ws[kk * PFN_OUTC + n];
            B[kt][nt].y = Ws[(kk + 1) * PFN_OUTC + n];
        }
    }

    // ---- A fragments (16x4 per tile) from LDS ----
    v2f A[2][3];
    #pragma unroll
    for (int mt = 0; mt < 2; ++mt) {
        const int q = 16 * mt + nlo;
        #pragma unroll
        for (int kt = 0; kt < 3; ++kt) {
            const int kk = 4 * kt + khalf;
            A[mt][kt].x = fsh[wv][q][kk];
            A[mt][kt].y = fsh[wv][q][kk + 1];
        }
    }

    // ---- WMMA: 2x4 C tiles, 3 K-steps each ----
    v8f C[2][4];
    #pragma unroll
    for (int mt = 0; mt < 2; ++mt)
        #pragma unroll
        for (int nt = 0; nt < 4; ++nt) {
            v8f c = {};
            #pragma unroll
            for (int kt = 0; kt < 3; ++kt) {
                c = __builtin_amdgcn_wmma_f32_16x16x4_f32(
                        /*neg_a=*/false, A[mt][kt],
                        /*neg_b=*/false, B[kt][nt],
                        /*c_mod=*/(short)0, c,
                        /*reuse_a=*/false, /*reuse_b=*/false);
            }
            C[mt][nt] = c;
        }

    // ---- epilogue: max over 32 points per column, +bias, relu, store ----
    #pragma unroll
    for (int nt = 0; nt < 4; ++nt) {
        float mv = -INFINITY;
        #pragma unroll
        for (int mt = 0; mt < 2; ++mt)
            #pragma unroll
            for (int r = 0; r < 8; ++r)
                mv = fmaxf(mv, C[mt][nt][r]);
        mv = fmaxf(mv, __shfl_xor(mv, 16, 32));   // combine rows r and r+8 halves
        const int n = nlo + 16 * nt;
        float y = fmaxf(mv + bias[n], 0.0f);
        if (active && lane < 16)
            out[(size_t)m * PFN_OUTC + n] = y;
    }
}

// ---------------------------------------------------------------------------
extern "C" void kernel_launch(void* const* d_in, const int* in_sizes, int n_in,
                              void* d_out, int out_size, void* d_ws, size_t ws_size,
                              hipStream_t stream) {
    const float* voxels = (const float*)d_in[0];   // (M,32,4) f32
    const int*   coords = (const int*)  d_in[1];   // (M,4)    i32
    const int*   nump   = (const int*)  d_in[2];   // (M,)     i32
    const float* W      = (const float*)d_in[3];   // (9,64)   f32
    const float* gamma  = (const float*)d_in[4];   // (64,)    f32
    const float* beta   = (const float*)d_in[5];   // (64,)    f32
    float* out = (float*)d_out;                    // (M,64)   f32

    const int Mn = in_sizes[0] / (PFN_P * 4);      // 40000

    float* S    = (float*)d_ws;                    // 64  floats: stats accumulators
    float* Ws   = S + 64;                          // 768 floats: BN-folded W (12x64)
    float* bias = Ws + 12 * PFN_OUTC;              // 64  floats: per-channel bias

    pfn_zero<<<1, 64, 0, stream>>>(S);
    pfn_stats<<<400, 256, 0, stream>>>(voxels, coords, nump, S, Mn);
    pfn_bn<<<1, 64, 0, stream>>>(S, W, gamma, beta, Ws, bias, Mn);
    pfn_main<<<(Mn + 7) / 8, 256, 0, stream>>>(voxels, coords, nump, Ws, bias, out, Mn);
}